// FilterTopKDetections_1554778161173
// MI455X (gfx1250) — compile-verified
//
#include <hip/hip_runtime.h>
#include <stdint.h>

#define B_ 8
#define A_ 120000
#define C_ 80
#define K_ 1000
#define NTHREADS 256
#define TILE 4096
#define NBINS 2048
#define CAND_MAX 4096
#define SIDE_MAX 8192

typedef uint32_t u32;
typedef uint64_t u64;
typedef __attribute__((ext_vector_type(4))) u32 u32x4;
typedef __attribute__((ext_vector_type(8))) u32 u32x8;

__device__ __forceinline__ u32 fbits(float f) { u32 u; __builtin_memcpy(&u, &f, 4); return u; }

// Build a 2D TDM descriptor: tile of `width` contiguous elements per row,
// `trows` rows, row pitch `stride` elements, 4-byte elements.
__device__ __forceinline__ void tdm_desc(u32x4& g0, u32x8& g1,
                                         const void* gptr, const void* lptr,
                                         u32 width, u32 rows, u32 stride, u32 trows) {
    u64 gaddr = (u64)(uintptr_t)gptr;
    g0 = u32x4{
        1u,                                             // count=1, user mode, no gather
        (u32)(uintptr_t)lptr,                           // lds_addr (low 32b of flat = LDS offset)
        (u32)gaddr,                                     // global_addr[31:0]
        (u32)((gaddr >> 32) & 0x01FFFFFFu) | (2u << 30) // global_addr[56:32] | type=2
    };
    g1 = u32x8{
        (2u << 16),                                     // data_size = 4 bytes
        (width & 0xFFFFu) << 16,                        // tensor_dim0[15:0]
        ((width >> 16) & 0xFFFFu) | ((rows & 0xFFFFu) << 16),  // tdim0[31:16] | tdim1[15:0]
        ((rows >> 16) & 0xFFFFu) | ((width & 0xFFFFu) << 16),  // tdim1[31:16] | tile_dim0
        (trows & 0xFFFFu),                              // tile_dim1 (tile_dim2=0)
        stride,                                         // tensor_dim0_stride[31:0]
        0u, 0u
    };
}

__device__ __forceinline__ void tdm_load(const void* g, const void* l,
                                         u32 width, u32 rows, u32 stride, u32 trows) {
    u32x4 g0; u32x8 g1;
    tdm_desc(g0, g1, g, l, width, rows, stride, trows);
    asm volatile("tensor_load_to_lds %0, %1" :: "s"(g0), "s"(g1) : "memory");
}

__device__ __forceinline__ void tdm_store(const void* g, const void* l,
                                          u32 width, u32 rows, u32 stride, u32 trows) {
    u32x4 g0; u32x8 g1;
    tdm_desc(g0, g1, g, l, width, rows, stride, trows);
    asm volatile("tensor_store_from_lds %0, %1" :: "s"(g0), "s"(g1) : "memory");
}

// Stream one class column (stride C_) through double-buffered TDM tiles, applying
// fn(anchor_index, value) to every element.
template <typename F>
__device__ __forceinline__ void stream_col(const float* col, float (&stage)[2][TILE],
                                           int tid, F&& fn) {
    const int nt = (A_ + TILE - 1) / TILE;
    if (tid < 32) {
        int t0 = A_ < TILE ? A_ : TILE;
        tdm_load(col, &stage[0][0], 1u, (u32)A_, (u32)C_, (u32)t0);
    }
    for (int t = 0; t < nt; ++t) {
        const int a0 = t * TILE;
        const int trows = (A_ - a0) < TILE ? (A_ - a0) : TILE;
        if (tid < 32) {
            if (t + 1 < nt) {
                const int a1 = a0 + TILE;
                const int r1 = A_ - a1;
                tdm_load(col + (size_t)a1 * C_, &stage[(t + 1) & 1][0],
                         1u, (u32)r1, (u32)C_, (u32)(r1 < TILE ? r1 : TILE));
                __builtin_amdgcn_s_wait_tensorcnt(1);   // previous tile landed
            } else {
                __builtin_amdgcn_s_wait_tensorcnt(0);
            }
        }
        __syncthreads();
        const float* buf = &stage[t & 1][0];
        for (int i = tid; i < trows; i += NTHREADS) fn(a0 + i, buf[i]);
        __syncthreads();                                 // readers done before next DMA overwrite
    }
}

__global__ __launch_bounds__(NTHREADS) void topk_filter_kernel(
    const float* __restrict__ scores,   // [B, A, C]
    const float* __restrict__ boxes,    // [B, A, 4]
    float* __restrict__ out)            // vals [B,K,C] ++ boxes_out [B,K,C,4]
{
    __shared__ float stage[2][TILE];    // 32 KB TDM landing buffers
    __shared__ u32   hist[NBINS];       // 8 KB
    __shared__ u64   side[SIDE_MAX];    // 64 KB: bucket-T1 elements from pass 2
    __shared__ u64   cand[CAND_MAX];    // 32 KB final sort keys
    __shared__ u32   sh_T1, sh_T2, sh_cnt, sh_side;

    const int blk = blockIdx.x;
    const int b = blk / C_;
    const int c = blk % C_;             // c fastest -> co-resident blocks share L2 lines
    const int tid = threadIdx.x;

    const float* col = scores + (size_t)b * A_ * C_ + c;

    // ---------------- pass 1: coarse histogram of bits[30:20] ----------------
    for (int i = tid; i < NBINS; i += NTHREADS) hist[i] = 0;
    __syncthreads();

    stream_col(col, stage, tid, [&](int /*a*/, float v) {
        atomicAdd(&hist[(fbits(v) >> 20) & (NBINS - 1)], 1u);
    });

    if (tid == 0) {
        u32 cum = 0; int t = NBINS - 1;
        for (; t >= 0; --t) { u32 h = hist[t]; if (cum + h >= (u32)K_) break; cum += h; }
        sh_T1 = (u32)(t < 0 ? 0 : t);
    }
    __syncthreads();
    const u32 T1 = sh_T1;
    __syncthreads();

    // ---------------- pass 2: refine + collect ----------------
    for (int i = tid; i < NBINS; i += NTHREADS) hist[i] = 0;
    if (tid == 0) { sh_cnt = 0; sh_side = 0; }
    __syncthreads();

    stream_col(col, stage, tid, [&](int a, float v) {
        const u32 ub = fbits(v);
        const u32 k1 = ub >> 20;
        const u64 key = ((u64)ub << 32) | (u32)(~(u32)a);  // ~idx: smaller idx wins ties
        if (k1 > T1) {
            u32 p = atomicAdd(&sh_cnt, 1u);
            if (p < CAND_MAX) cand[p] = key;               // definitely in top-K
        } else if (k1 == T1) {
            atomicAdd(&hist[(ub >> 9) & (NBINS - 1)], 1u);
            u32 p = atomicAdd(&sh_side, 1u);
            if (p < SIDE_MAX) side[p] = key;               // decided after T2 known
        }
    });

    if (tid == 0) {
        const u32 cgt = sh_cnt;                            // elements strictly above T1 (< K)
        const u32 need = ((u32)K_ > cgt) ? ((u32)K_ - cgt) : 1u;
        u32 cum = 0; int t = NBINS - 1;
        for (; t >= 0; --t) { u32 h = hist[t]; if (cum + h >= need) break; cum += h; }
        sh_T2 = (u32)(t < 0 ? 0 : t);
    }
    __syncthreads();
    const u32 T2 = sh_T2;
    const u32 ns = sh_side < SIDE_MAX ? sh_side : SIDE_MAX;
    for (u32 i = tid; i < ns; i += NTHREADS) {
        const u64 key = side[i];
        if ((((u32)(key >> 32) >> 9) & (NBINS - 1)) >= T2) {
            u32 p = atomicAdd(&sh_cnt, 1u);
            if (p < CAND_MAX) cand[p] = key;
        }
    }
    __syncthreads();
    u32 cnt = sh_cnt; if (cnt > CAND_MAX) cnt = CAND_MAX;
    for (u32 i = cnt + tid; i < CAND_MAX; i += NTHREADS) cand[i] = 0ull;
    __syncthreads();

    // ---------------- bitonic sort, descending, 4096 keys in LDS ----------------
    for (u32 ksz = 2; ksz <= CAND_MAX; ksz <<= 1) {
        for (u32 j = ksz >> 1; j > 0; j >>= 1) {
            for (u32 i = tid; i < CAND_MAX; i += NTHREADS) {
                const u32 l = i ^ j;
                if (l > i) {
                    const u64 x = cand[i], y = cand[l];
                    const bool asc = ((i & ksz) == 0);
                    if (asc ? (x < y) : (x > y)) { cand[i] = y; cand[l] = x; }
                }
            }
            __syncthreads();
        }
    }

    // ---------------- stage outputs in LDS, emit via TDM strided stores ----------------
    float* valbuf = &stage[0][0];                          // K floats, contiguous
    float* boxbuf = &stage[1][0];                          // K*4 floats, contiguous
    for (int kk = tid; kk < K_; kk += NTHREADS) {
        const u64 key = cand[kk];
        const u32 vb  = (u32)(key >> 32);
        u32 idx = ~(u32)key;
        if (idx >= (u32)A_) idx = 0;                       // degenerate underflow padding only
        float v; __builtin_memcpy(&v, &vb, 4);
        valbuf[kk] = v;
        const float4 bx = *(const float4*)(boxes + ((size_t)b * A_ + idx) * 4);
        boxbuf[kk * 4 + 0] = bx.x; boxbuf[kk * 4 + 1] = bx.y;
        boxbuf[kk * 4 + 2] = bx.z; boxbuf[kk * 4 + 3] = bx.w;
    }
    __syncthreads();

    if (tid < 32) {
        const size_t VOFF = (size_t)B_ * K_ * C_;
        // vals[b, kk, c]: 1 float per row, K rows, row pitch C
        tdm_store(out + (size_t)b * K_ * C_ + c, valbuf, 1u, (u32)K_, (u32)C_, (u32)K_);
        // boxes_out[b, kk, c, 0:4]: 4 floats per row, K rows, row pitch 4*C
        tdm_store(out + VOFF + ((size_t)b * K_ * C_ + c) * 4, boxbuf,
                  4u, (u32)K_, (u32)(4 * C_), (u32)K_);
        __builtin_amdgcn_s_wait_tensorcnt(0);
    }
}

extern "C" void kernel_launch(void* const* d_in, const int* in_sizes, int n_in,
                              void* d_out, int out_size, void* d_ws, size_t ws_size,
                              hipStream_t stream) {
    (void)in_sizes; (void)n_in; (void)d_ws; (void)ws_size; (void)out_size;
    const float* scores = (const float*)d_in[0];
    const float* boxes  = (const float*)d_in[1];
    float* out = (float*)d_out;
    hipLaunchKernelGGL(topk_filter_kernel, dim3(B_ * C_), dim3(NTHREADS), 0, stream,
                       scores, boxes, out);
}